// SpecialSparseConv_21698174779788
// MI455X (gfx1250) — compile-verified
//
#include <hip/hip_runtime.h>

#define N_IN   100000
#define N_OUT  100000
#define E_TOT  3200000
#define KK     55
#define CIN    32
#define COUT   32
#define ROWS_PER_BLOCK 16
#define KDIM   (KK * CIN)        /* 1760 : contraction length            */
#define S_BUCK 1764              /* padded LDS row stride (bank-conflict-free) */
#define NWAVES 8
#define BLOCK  (NWAVES * 32)
#define NSTEPS (KDIM / 4)        /* 440 WMMA K-steps                     */
#define PACK_ELEMS (NSTEPS * 32 * 4)   /* packed-B floats = 56320        */

typedef __attribute__((ext_vector_type(2))) float v2f;
typedef __attribute__((ext_vector_type(4))) float v4f;
typedef __attribute__((ext_vector_type(8))) float v8f;

// Pre-pass: repack kernel matrix (1760 x 32) into WMMA B-fragment order so the
// hot loop does one coalesced global_load_b128 per K-step per lane.
__global__ void __launch_bounds__(256)
pack_b_kernel(const float* __restrict__ kern, float* __restrict__ packed) {
    const int t = blockIdx.x * 256 + threadIdx.x;
    if (t >= NSTEPS * 32) return;
    const int step = t >> 5;
    const int lane = t & 31;
    const int kSel = (lane >> 4) << 1;       // lanes 0-15 -> K rows +0/+1, 16-31 -> +2/+3
    const int n    = lane & 15;
    const int kk0  = step * 4;
    v4f b;
    b.x = kern[(kk0 + kSel) * COUT + n];          // N cols 0..15, K row kSel
    b.y = kern[(kk0 + kSel + 1) * COUT + n];      //               K row kSel+1
    b.z = kern[(kk0 + kSel) * COUT + n + 16];     // N cols 16..31
    b.w = kern[(kk0 + kSel + 1) * COUT + n + 16];
    *((v4f*)packed + t) = b;
}

// Fused sparse-conv: bucketize edges into LDS, then WMMA f32 16x16x4 contraction.
// One workgroup = 16 output rows (the WMMA M dimension). 8 waves of 32.
template <bool PACKED>
__global__ void __launch_bounds__(BLOCK)
sparse_conv_wmma_kernel(const float* __restrict__ feat,       // [N_IN, CIN]
                        const int* __restrict__ nidx,         // [E]
                        const int* __restrict__ nk,           // [E]
                        const long long* __restrict__ splits, // [N_OUT+1] int64
                        const float* __restrict__ imp,        // [N_IN]
                        const float* __restrict__ kern,       // [KDIM, COUT]
                        const float* __restrict__ kernPacked, // [NSTEPS*32*4] fragment order
                        const float* __restrict__ bias,       // [COUT]
                        float* __restrict__ out)              // [N_OUT*COUT] ++ [N_OUT]
{
    extern __shared__ char smem[];
    float* bucket = (float*)smem;                       // 16 * 1764 f32 = 112896 B
    float* impL   = bucket + ROWS_PER_BLOCK * S_BUCK;   // 16 f32
    float* red    = impL + ROWS_PER_BLOCK;              // 8 * 512 f32 = 16384 B
    int*   sInt   = (int*)(red + NWAVES * 512);         // 17 ints

    const int tid     = threadIdx.x;
    const int lane    = tid & 31;
    const int wave    = tid >> 5;
    const int rowBase = blockIdx.x * ROWS_PER_BLOCK;

    // ---- zero-init LDS ----
    for (int i = tid; i < ROWS_PER_BLOCK * S_BUCK; i += BLOCK) bucket[i] = 0.0f;
    if (tid < ROWS_PER_BLOCK)  impL[tid] = 0.0f;
    if (tid <= ROWS_PER_BLOCK) sInt[tid] = (int)splits[rowBase + tid];
    __syncthreads();

    // ---- Stage A: gather + scale + bucketize (lane == channel) ----
    for (int r = 0; r < ROWS_PER_BLOCK; ++r) {
        const int e0 = sInt[r];
        const int e1 = sInt[r + 1];
        float wsum = 0.0f;                       // wave-uniform importance accumulator
        for (int e = e0 + wave; e < e1; e += NWAVES) {
            const int   src = nidx[e];           // wave-uniform
            const int   k   = nk[e];             // wave-uniform
            const float w   = imp[src];
            const float x   = feat[src * CIN + lane] * w;   // coalesced 128B
            wsum += w;
            __hip_atomic_fetch_add(&bucket[r * S_BUCK + k * CIN + lane], x,
                                   __ATOMIC_RELAXED, __HIP_MEMORY_SCOPE_WORKGROUP);
        }
        if (lane == 0 && wsum != 0.0f)           // one LDS atomic per row per wave
            __hip_atomic_fetch_add(&impL[r], wsum,
                                   __ATOMIC_RELAXED, __HIP_MEMORY_SCOPE_WORKGROUP);
    }
    __syncthreads();

    // ---- Stage B: D = A(16x1760) * B(1760x32), K-range split over 8 waves ----
    v8f c0 = {};   // N cols 0..15
    v8f c1 = {};   // N cols 16..31
    const int m    = lane & 15;          // A row M (lanes 16-31 mirror with K+2/K+3)
    const int kSel = (lane >> 4) << 1;

#pragma unroll 5
    for (int i = 0; i < NSTEPS / NWAVES; ++i) {   // 55 iterations per wave
        const int step = wave + NWAVES * i;       // wave-uniform -> EXEC stays full
        const int kk0  = step * 4;
        // A fragment from LDS buckets (padded stride -> 64 distinct banks)
        v2f a;
        a.x = bucket[m * S_BUCK + kk0 + kSel];
        a.y = bucket[m * S_BUCK + kk0 + kSel + 1];
        v2f b0, b1;
        if (PACKED) {
            const v4f b = *((const v4f*)kernPacked + step * 32 + lane); // 1x b128, coalesced
            b0.x = b.x; b0.y = b.y;
            b1.x = b.z; b1.y = b.w;
        } else {
            const float* kp = kern + (kk0 + kSel) * COUT + m;
            b0.x = kp[0];   b0.y = kp[COUT];
            b1.x = kp[16];  b1.y = kp[COUT + 16];
        }
        c0 = __builtin_amdgcn_wmma_f32_16x16x4_f32(false, a, false, b0, (short)0, c0, false, false);
        c1 = __builtin_amdgcn_wmma_f32_16x16x4_f32(false, a, false, b1, (short)0, c1, false, false);
    }

    // ---- cross-wave reduction of partial C tiles through LDS ----
#pragma unroll
    for (int i = 0; i < 8; ++i) {
        red[wave * 512 + i * 32 + lane]       = c0[i];
        red[wave * 512 + (8 + i) * 32 + lane] = c1[i];
    }
    __syncthreads();

    if (wave == 0) {
#pragma unroll
        for (int i = 0; i < 16; ++i) {
            float v = 0.0f;
#pragma unroll
            for (int w2 = 0; w2 < NWAVES; ++w2)
                v += red[w2 * 512 + i * 32 + lane];
            // un-permute the WMMA C layout: VGPR i, lane l -> (M, N)
            const int mm = (i & 7) + ((lane >> 4) << 3);
            const int nn = (lane & 15) + ((i >= 8) ? 16 : 0);
            const float norm = impL[mm];
            if (norm != 0.0f) v /= norm;
            v += bias[nn];
            out[(size_t)(rowBase + mm) * COUT + nn] = v;
        }
        if (lane < ROWS_PER_BLOCK)
            out[(size_t)N_OUT * COUT + rowBase + lane] = impL[lane];
    }
}

extern "C" void kernel_launch(void* const* d_in, const int* in_sizes, int n_in,
                              void* d_out, int out_size, void* d_ws, size_t ws_size,
                              hipStream_t stream) {
    (void)in_sizes; (void)n_in; (void)out_size;
    const float*     feat   = (const float*)d_in[0];
    const int*       nidx   = (const int*)d_in[1];
    const int*       nk     = (const int*)d_in[2];
    const long long* splits = (const long long*)d_in[3];   // int64 row splits
    const float*     imp    = (const float*)d_in[4];
    const float*     kern   = (const float*)d_in[5];
    const float*     bias   = (const float*)d_in[6];
    float*           out    = (float*)d_out;

    const size_t lds = (size_t)(ROWS_PER_BLOCK * S_BUCK + ROWS_PER_BLOCK + NWAVES * 512) * sizeof(float)
                     + (ROWS_PER_BLOCK + 1) * sizeof(int);
    const int nBlocks = N_OUT / ROWS_PER_BLOCK;   // 6250

    const bool usePacked = (d_ws != nullptr) && (ws_size >= (size_t)PACK_ELEMS * sizeof(float));
    if (usePacked) {
        float* packed = (float*)d_ws;
        pack_b_kernel<<<(NSTEPS * 32 + 255) / 256, 256, 0, stream>>>(kern, packed);
        sparse_conv_wmma_kernel<true><<<nBlocks, BLOCK, lds, stream>>>(
            feat, nidx, nk, splits, imp, kern, packed, bias, out);
    } else {
        sparse_conv_wmma_kernel<false><<<nBlocks, BLOCK, lds, stream>>>(
            feat, nidx, nk, splits, imp, kern, nullptr, bias, out);
    }
}